// UpdateNetV2_86449101734004
// MI455X (gfx1250) — compile-verified
//
#include <hip/hip_runtime.h>
#include <math.h>

#define B_ 4
#define N_ 1024
#define H_ 256
#define D_ 32

typedef __attribute__((ext_vector_type(2))) float v2f;
typedef __attribute__((ext_vector_type(8))) float v8f;

__device__ __forceinline__ float silu_f(float x) {
    return x / (1.0f + __expf(-x));
}
// numerically stable log(sigmoid(x)) = min(x,0) - log1p(exp(-|x|))
__device__ __forceinline__ float logsig_f(float x) {
    return fminf(x, 0.0f) - log1pf(__expf(-fabsf(x)));
}

// ---------------------------------------------------------------------------
// Kernel 1: projection GEMM on the WMMA pipe (f32, 16x16x4).
// C(4096 x 864) = token(4096 x 256) @ [W_kv | W_kvh | W_decay | W_decayh].
// One wave per 16x16 output tile; all segment boundaries (64/576/608) are
// 16-aligned so a tile never straddles two weight matrices.
// A frag: A[M=l16, K=k0+2*half+{0,1}] from token rows (64-bit load per lane).
// B frag: B[K, N=l16] -> two 16-wide coalesced rows of W.
// ---------------------------------------------------------------------------
__global__ __launch_bounds__(32) void proj_wmma_kernel(
    const float* __restrict__ token,
    const float* __restrict__ Wkv,  const float* __restrict__ Wkvh,
    const float* __restrict__ Wdec, const float* __restrict__ Wdech,
    float* __restrict__ kv,  float* __restrict__ kvh,
    float* __restrict__ dec, float* __restrict__ dech)
{
    const int  mt  = blockIdx.x;            // M tile: rows mt*16 .. mt*16+15 of (B*N)
    const int  c0  = blockIdx.y * 16;       // column in concatenated 864-wide output
    const int  lane = threadIdx.x;
    const int  half = lane >> 4;            // 0: lanes 0-15, 1: lanes 16-31
    const int  l16  = lane & 15;

    const float* W; float* dst; int width, col;
    if (c0 < 64)       { W = Wkv;   width = 64;  col = c0;       dst = kv;   }
    else if (c0 < 576) { W = Wkvh;  width = 512; col = c0 - 64;  dst = kvh;  }
    else if (c0 < 608) { W = Wdec;  width = 32;  col = c0 - 576; dst = dec;  }
    else               { W = Wdech; width = 256; col = c0 - 608; dst = dech; }

    const long  row0 = (long)mt * 16;
    const float* tokA = token + (row0 + l16) * H_;   // this lane's token row

    v8f acc = {0.f, 0.f, 0.f, 0.f, 0.f, 0.f, 0.f, 0.f};

    #pragma unroll 8
    for (int k0 = 0; k0 < H_; k0 += 4) {
        v2f a, bb;
        const int ka = k0 + 2 * half;
        a.x  = tokA[ka];
        a.y  = tokA[ka + 1];
        bb.x = W[(size_t)ka * width + col + l16];
        bb.y = W[(size_t)(ka + 1) * width + col + l16];
        acc = __builtin_amdgcn_wmma_f32_16x16x4_f32(
            false, a, false, bb, (short)0, acc, false, false);
    }

    #pragma unroll
    for (int r = 0; r < 8; ++r)
        dst[(size_t)(row0 + r + 8 * half) * width + col + l16] = acc[r];
}

// ---------------------------------------------------------------------------
// Kernel 2: per-(b,h) fused gate/decay/scan + f32 WMMA contraction.
//   wm[d,e] = sum_n silu(k[n,d]*kh[n]) * exp(S[n,d]) * silu(v[n,e]*vh[n])
// with S[n,d] = inclusive cumsum over n of ld_rev, ld_rev[0]=0,
// ld_rev[n] = log_sigmoid(dec[N-n,d]*dech[N-n])  (reference flip+cumsum).
// 4 waves / block; wave w owns output tile (di,ei) = (w>>1, w&1) * 16.
// ---------------------------------------------------------------------------
__global__ __launch_bounds__(128) void wm_kernel(
    const float* __restrict__ kv,  const float* __restrict__ kvh,
    const float* __restrict__ dec, const float* __restrict__ dech,
    float* __restrict__ out)
{
    __shared__ float s_s[32][D_];       // chunk of inclusive cumsum S[n_local][d]
    __shared__ float a_s[32][D_ + 1];   // k_scaled[n_local][d]   (A^T source)
    __shared__ float v_s[32][D_ + 1];   // v[n_local][e]          (B source)

    const int bh   = blockIdx.x;
    const int b    = bh >> 8;          // / H_
    const int h    = bh & (H_ - 1);
    const int tid  = threadIdx.x;
    const int lane = tid & 31;
    const int wave = tid >> 5;
    const int half = lane >> 4;        // 0: lanes 0-15, 1: lanes 16-31
    const int l16  = lane & 15;
    const int di   = (wave >> 1) * 16; // output tile row base (d)
    const int ei   = (wave & 1) * 16;  // output tile col base (e)

    v8f acc = {0.f, 0.f, 0.f, 0.f, 0.f, 0.f, 0.f, 0.f};
    float run = 0.0f;                  // wave0 lane d: running cumsum, persists across chunks

    const long rowbase = (long)b * N_;

    for (int n0 = 0; n0 < N_; n0 += 32) {
        // ---- Prefetch next chunk's gate inputs toward this WGP (global_prefetch_b8)
        if (n0 + 32 < N_) {
            const long rn = rowbase + n0 + 32 + (tid >> 2);     // 32 rows x 4 threads
            __builtin_prefetch(&kv[rn * 64 + (tid & 3) * 16], 0, 1);
            if (tid < 32)
                __builtin_prefetch(&kvh[(rowbase + n0 + 32 + tid) * 512 + h], 0, 1);
        }

        // ---- Stage 1: extend suffix cumsum of log-decay (wave 0, lane = d) ----
        if (wave == 0) {
            for (int j = 0; j < 32; ++j) {
                const int n = n0 + j;
                float ld = 0.0f;
                if (n != 0) {
                    const long r = rowbase + (N_ - n);     // reversed index
                    const float x = dec[r * D_ + lane] * dech[r * H_ + h];
                    ld = logsig_f(x);
                }
                run += ld;
                s_s[j][lane] = run;                        // inclusive cumsum
            }
        }
        __syncthreads();

        // ---- Stage 2: build gated/scaled tiles in LDS ----
        for (int idx = tid; idx < 32 * D_; idx += 128) {
            const int j = idx >> 5, d = idx & 31;
            const long r = rowbase + n0 + j;
            const float kh = kvh[r * 512 + h];
            const float vh = kvh[r * 512 + 256 + h];
            const float kx = silu_f(kv[r * 64 + d]      * kh);
            const float vx = silu_f(kv[r * 64 + 32 + d] * vh);
            a_s[j][d] = kx * __expf(s_s[j][d]);
            v_s[j][d] = vx;
        }
        __syncthreads();

        // ---- Stage 3: acc[d,e] += sum_j A^T[d,j] * V[j,e]  (f32 WMMA, K=4) ----
        #pragma unroll
        for (int k0 = 0; k0 < 32; k0 += 4) {
            v2f a, bb;
            a.x  = a_s[k0 + 2 * half + 0][di + l16];
            a.y  = a_s[k0 + 2 * half + 1][di + l16];
            bb.x = v_s[k0 + 2 * half + 0][ei + l16];
            bb.y = v_s[k0 + 2 * half + 1][ei + l16];
            acc = __builtin_amdgcn_wmma_f32_16x16x4_f32(
                false, a, false, bb, (short)0, acc, false, false);
        }
        __syncthreads();
    }

    // ---- Store: C/D layout -> out[b, d*D + e, h]
    #pragma unroll
    for (int r = 0; r < 8; ++r) {
        const int d = di + r + 8 * half;
        const int e = ei + l16;
        out[(((long)b * (D_ * D_)) + d * D_ + e) * H_ + h] = acc[r];
    }
}

// ---------------------------------------------------------------------------
extern "C" void kernel_launch(void* const* d_in, const int* in_sizes, int n_in,
                              void* d_out, int out_size, void* d_ws, size_t ws_size,
                              hipStream_t stream)
{
    const float* token = (const float*)d_in[0];   // (B,N,H)
    const float* Wkv   = (const float*)d_in[1];   // (H,2D)
    const float* Wkvh  = (const float*)d_in[2];   // (H,2H)
    const float* Wdec  = (const float*)d_in[3];   // (H,D)
    const float* Wdech = (const float*)d_in[4];   // (H,H)

    float* ws   = (float*)d_ws;
    float* kv   = ws;                                   // B*N*64
    float* kvh  = kv   + (size_t)B_ * N_ * 64;          // B*N*512
    float* dec  = kvh  + (size_t)B_ * N_ * 512;         // B*N*32
    float* dech = dec  + (size_t)B_ * N_ * 32;          // B*N*256  (~13.5 MB total)

    dim3 pgrid(B_ * N_ / 16, 864 / 16);                 // 256 x 54 tiles
    proj_wmma_kernel<<<pgrid, 32, 0, stream>>>(token, Wkv, Wkvh, Wdec, Wdech,
                                               kv, kvh, dec, dech);
    wm_kernel<<<B_ * H_, 128, 0, stream>>>(kv, kvh, dec, dech, (float*)d_out);
}